// GraphSage_90692529422657
// MI455X (gfx1250) — compile-verified
//
#include <hip/hip_runtime.h>
#include <math.h>

// Problem constants (match reference)
#define NNODES 50000
#define NEDGES 800000
#define FDIM   64
#define HDIM   64
#define CDIM   40
#define CPAD   48   // padded class dim (3 x 16-col WMMA tiles)

typedef __attribute__((ext_vector_type(2))) float v2f;
typedef __attribute__((ext_vector_type(8))) float v8f;

// ---------------------------------------------------------------------------
// One 16x16 output tile accumulation: acc += A[16x64] @ B[64 x ldb](cols col0..col0+15)
// A is row-major with lda=64 (caller passes A + row0*64).
// B is row-major [64, ldb]; columns >= ncol are read as zero (guard for C=40).
// Uses V_WMMA_F32_16X16X4_F32: A 16x4 (v2f/lane), B 4x16 (v2f/lane), C/D 16x16 (v8f/lane).
// A layout: lane L -> row m=L&15, k-pair kb=(L>>4)*2 : {A[m][k+kb], A[m][k+kb+1]}
// B layout (symmetric): lane L -> col n=L&15, kb=(L>>4)*2 : {B[k+kb][n], B[k+kb+1][n]}
// ---------------------------------------------------------------------------
__device__ __forceinline__ v8f wmma_tile_accum(const float* __restrict__ A,
                                               const float* __restrict__ B,
                                               int ldb, int col0, int ncol, v8f acc)
{
    const int lane = threadIdx.x & 31;
    const int m    = lane & 15;
    const int kh   = (lane >> 4) << 1;          // 0 or 2
    const int col  = col0 + m;
    const bool cv  = (col < ncol);
    const int cs   = cv ? col : 0;
#pragma unroll
    for (int k = 0; k < 64; k += 4) {
        v2f a = *(const v2f*)(A + m * 64 + k + kh);          // 8B aligned
        v2f b;
        const float* bp = B + (k + kh) * ldb + cs;
        b.x = cv ? bp[0]   : 0.0f;
        b.y = cv ? bp[ldb] : 0.0f;
        acc = __builtin_amdgcn_wmma_f32_16x16x4_f32(
                  /*neg_a=*/false, a, /*neg_b=*/false, b,
                  /*c_mod=*/(short)0, acc, /*reuse_a=*/false, /*reuse_b=*/false);
    }
    return acc;
}

// Store 16x16 tile: D layout lane L -> col n=L&15, VGPR v -> row v + (L>>4)*8
__device__ __forceinline__ void store_tile16(float* __restrict__ O, int ldo,
                                             int row0, int col0,
                                             const float* __restrict__ bias, int nbias,
                                             bool do_relu, v8f acc)
{
    const int lane = threadIdx.x & 31;
    const int n    = lane & 15;
    const int mo   = (lane >> 4) << 3;          // 0 or 8
    const int col  = col0 + n;
    const float bv = (col < nbias) ? bias[col] : 0.0f;
#pragma unroll
    for (int v = 0; v < 8; ++v) {
        float val = acc[v] + bv;
        if (do_relu) val = fmaxf(val, 0.0f);
        O[(size_t)(row0 + mo + v) * ldo + col] = val;
    }
}

// ---------------------------------------------------------------------------
// Y[N,64] = relu(X[N,64] @ W[64,64] + bias[64])   (shared pool transform)
// one wave per 16x16 tile; tiles = (N/16) * 4
// ---------------------------------------------------------------------------
__global__ void relu_linear64_kernel(const float* __restrict__ X,
                                     const float* __restrict__ W,
                                     const float* __restrict__ bias,
                                     float* __restrict__ Y)
{
    const int wid    = (blockIdx.x * blockDim.x + threadIdx.x) >> 5;
    const int nTiles = (NNODES / 16) * (HDIM / 16);
    if (wid >= nTiles) return;                  // uniform per wave -> EXEC all-1s
    const int row0 = (wid >> 2) * 16;
    const int col0 = (wid & 3) * 16;
    v8f acc = {};
    acc = wmma_tile_accum(X + (size_t)row0 * 64, W, 64, col0, 64, acc);
    store_tile16(Y, 64, row0, col0, bias, 64, /*relu=*/true, acc);
}

// ---------------------------------------------------------------------------
// H[N,64] = relu(AGG[N,64] @ Wl[64,64] + X[N,64] @ Wr[64,64] + bias[64])
// ---------------------------------------------------------------------------
__global__ void sage_combine64_kernel(const float* __restrict__ AGG,
                                      const float* __restrict__ Wl,
                                      const float* __restrict__ X,
                                      const float* __restrict__ Wr,
                                      const float* __restrict__ bias,
                                      float* __restrict__ H)
{
    const int wid    = (blockIdx.x * blockDim.x + threadIdx.x) >> 5;
    const int nTiles = (NNODES / 16) * (HDIM / 16);
    if (wid >= nTiles) return;
    const int row0 = (wid >> 2) * 16;
    const int col0 = (wid & 3) * 16;
    v8f acc = {};
    acc = wmma_tile_accum(AGG + (size_t)row0 * 64, Wl, 64, col0, 64, acc);
    acc = wmma_tile_accum(X   + (size_t)row0 * 64, Wr, 64, col0, 64, acc);
    store_tile16(H, 64, row0, col0, bias, 64, /*relu=*/true, acc);
}

// ---------------------------------------------------------------------------
// PRE[N,48] = AGG[N,64] @ Wl2[64,40] + H[N,64] @ Wr2[64,40] + b2[40]  (cols 40..47 junk)
// ---------------------------------------------------------------------------
__global__ void sage_out_kernel(const float* __restrict__ AGG,
                                const float* __restrict__ Wl2,
                                const float* __restrict__ H,
                                const float* __restrict__ Wr2,
                                const float* __restrict__ b2,
                                float* __restrict__ PRE)
{
    const int wid    = (blockIdx.x * blockDim.x + threadIdx.x) >> 5;
    const int nTiles = (NNODES / 16) * (CPAD / 16);
    if (wid >= nTiles) return;
    const int rowT = wid / (CPAD / 16);
    const int colT = wid % (CPAD / 16);
    const int row0 = rowT * 16;
    const int col0 = colT * 16;
    v8f acc = {};
    acc = wmma_tile_accum(AGG + (size_t)row0 * 64, Wl2, CDIM, col0, CDIM, acc);
    acc = wmma_tile_accum(H   + (size_t)row0 * 64, Wr2, CDIM, col0, CDIM, acc);
    store_tile16(PRE, CPAD, row0, col0, b2, CDIM, /*relu=*/false, acc);
}

// ---------------------------------------------------------------------------
// Edge segment-max: agg[dst] = max(agg[dst], y[src]) elementwise over 64 feats.
// y >= 0 (relu output), so uint bit-pattern atomicMax == float max; agg init 0
// also implements maximum(segment_max, 0) for isolated nodes exactly.
// 16 threads per edge, float4 per thread.
// ---------------------------------------------------------------------------
__global__ void edge_max_kernel(const float* __restrict__ y,
                                const int* __restrict__ src,
                                const int* __restrict__ dst,
                                unsigned int* __restrict__ agg)
{
    const int t = blockIdx.x * blockDim.x + threadIdx.x;
    const int e = t >> 4;
    if (e >= NEDGES) return;
    const int f = (t & 15) << 2;
    const int s = src[e];
    const int d = dst[e];
    const float4 v = *(const float4*)(y + (size_t)s * 64 + f);
    unsigned int* o = agg + (size_t)d * 64 + f;
    atomicMax(o + 0, __float_as_uint(v.x));
    atomicMax(o + 1, __float_as_uint(v.y));
    atomicMax(o + 2, __float_as_uint(v.z));
    atomicMax(o + 3, __float_as_uint(v.w));
}

// ---------------------------------------------------------------------------
// Row-wise log_softmax over 40 classes; input ld=48. One wave per row.
// ---------------------------------------------------------------------------
__global__ void log_softmax_kernel(const float* __restrict__ PRE,
                                   float* __restrict__ OUT)
{
    const int row = blockIdx.x * (blockDim.x >> 5) + (threadIdx.x >> 5);
    if (row >= NNODES) return;
    const int lane = threadIdx.x & 31;
    const float* r = PRE + (size_t)row * CPAD;
    const bool hi = (lane + 32 < CDIM);
    const float v0 = r[lane];                       // lane < 32 <= 40 always valid
    const float v1 = hi ? r[lane + 32] : -INFINITY;
    float m = fmaxf(v0, v1);
#pragma unroll
    for (int off = 16; off > 0; off >>= 1) m = fmaxf(m, __shfl_xor(m, off, 32));
    float s = __expf(v0 - m) + (hi ? __expf(v1 - m) : 0.0f);
#pragma unroll
    for (int off = 16; off > 0; off >>= 1) s += __shfl_xor(s, off, 32);
    const float lse = m + __logf(s);
    OUT[(size_t)row * CDIM + lane] = v0 - lse;
    if (hi) OUT[(size_t)row * CDIM + lane + 32] = v1 - lse;
}

// ---------------------------------------------------------------------------
extern "C" void kernel_launch(void* const* d_in, const int* in_sizes, int n_in,
                              void* d_out, int out_size, void* d_ws, size_t ws_size,
                              hipStream_t stream)
{
    const float* x      = (const float*)d_in[0];          // [N,64]
    const int*   eidx   = (const int*)  d_in[1];          // [2,E]
    const float* W_pool = (const float*)d_in[2];          // [64,64]
    const float* b_pool = (const float*)d_in[3];          // [64]
    const float* Wl1    = (const float*)d_in[4];          // [64,64]
    const float* Wr1    = (const float*)d_in[5];          // [64,64]
    const float* b1     = (const float*)d_in[6];          // [64]
    const float* Wl2    = (const float*)d_in[7];          // [64,40]
    const float* Wr2    = (const float*)d_in[8];          // [64,40]
    const float* b2     = (const float*)d_in[9];          // [40]
    float*       out    = (float*)d_out;                  // [N,40]

    const int* src = eidx;            // edge_index[0]
    const int* dst = eidx + NEDGES;   // edge_index[1]

    // Workspace layout (floats): y[N*64] | agg[N*64] | h[N*64] | pre[N*48]
    float* y   = (float*)d_ws;
    float* agg = y   + (size_t)NNODES * 64;
    float* h   = agg + (size_t)NNODES * 64;
    float* pre = h   + (size_t)NNODES * 64;

    const int gemm64Waves = (NNODES / 16) * (HDIM / 16);         // 12500
    const int gemm64Blks  = (gemm64Waves * 32 + 255) / 256;
    const int gemmCWaves  = (NNODES / 16) * (CPAD / 16);         // 9375
    const int gemmCBlks   = (gemmCWaves * 32 + 255) / 256;
    const int edgeBlks    = (NEDGES * 16 + 255) / 256;
    const int lsmBlks     = (NNODES + 7) / 8;                    // 8 rows / 256-thread block

    // ---- Layer 1 ----
    relu_linear64_kernel<<<gemm64Blks, 256, 0, stream>>>(x, W_pool, b_pool, y);
    hipMemsetAsync(agg, 0, (size_t)NNODES * 64 * sizeof(float), stream);
    edge_max_kernel<<<edgeBlks, 256, 0, stream>>>(y, src, dst, (unsigned int*)agg);
    sage_combine64_kernel<<<gemm64Blks, 256, 0, stream>>>(agg, Wl1, x, Wr1, b1, h);

    // ---- Layer 2 (dropout = identity in eval) ----
    relu_linear64_kernel<<<gemm64Blks, 256, 0, stream>>>(h, W_pool, b_pool, y);
    hipMemsetAsync(agg, 0, (size_t)NNODES * 64 * sizeof(float), stream);
    edge_max_kernel<<<edgeBlks, 256, 0, stream>>>(y, src, dst, (unsigned int*)agg);
    sage_out_kernel<<<gemmCBlks, 256, 0, stream>>>(agg, Wl2, h, Wr2, b2, pre);

    // ---- log_softmax ----
    log_softmax_kernel<<<lsmBlks, 256, 0, stream>>>(pre, out);
}